// OT_DPL_2087354106136
// MI455X (gfx1250) — compile-verified
//
#include <hip/hip_runtime.h>
#include <math.h>

#define B_   16
#define C_   256
#define HW_  4096
#define N_   5
#define E_   512
#define MCH_ 8              // m-chunks for K4 (4096/8 = 512 each)
#define SCALE_ 0.0625f      // 256^-0.5
#define EPS_   0.05f
#define TOL_   1e-4f

typedef __attribute__((ext_vector_type(2))) float v2f;
typedef __attribute__((ext_vector_type(8))) float v8f;

// ---------- shared helpers ----------
__device__ __forceinline__ float block_ln_row(float x, float* red, float s, float b_, int tid) {
    // LayerNorm over a row of C_=256 values, one value per thread. blockDim.x == 256.
    red[tid] = x; __syncthreads();
    for (int off = 128; off > 0; off >>= 1) { if (tid < off) red[tid] += red[tid + off]; __syncthreads(); }
    float mean = red[0] * (1.0f / 256.0f); __syncthreads();
    float d = x - mean;
    red[tid] = d * d; __syncthreads();
    for (int off = 128; off > 0; off >>= 1) { if (tid < off) red[tid] += red[tid + off]; __syncthreads(); }
    float var = red[0] * (1.0f / 256.0f); __syncthreads();
    return d * rsqrtf(var + 1e-5f) * s + b_;
}

// ---------- K0: batch-independent prompt pipeline (runs once, 1 block) ----------
__global__ void k0_prompt_pipeline(
    const float* __restrict__ prompts,
    const float* __restrict__ qw, const float* __restrict__ qb_,
    const float* __restrict__ kw, const float* __restrict__ kb_,
    const float* __restrict__ vw, const float* __restrict__ vb_,
    const float* __restrict__ ls, const float* __restrict__ lb,
    const float* __restrict__ wq_w, const float* __restrict__ wq_b,
    const float* __restrict__ wk_w, const float* __restrict__ wk_b,
    float* __restrict__ P1, float* __restrict__ Qk, float* __restrict__ qbo)
{
    __shared__ float sP[N_ * C_], sQ[N_ * C_], sK[N_ * C_], sV[N_ * C_], sQc[N_ * C_];
    __shared__ float sAttn[N_ * N_], red[C_];
    const int tid = threadIdx.x;

    for (int i = tid; i < N_ * C_; i += C_) sP[i] = prompts[i];
    __syncthreads();

    // Q,K,V = P @ W + b
    for (int i = tid; i < N_ * C_; i += C_) {
        int n = i >> 8, c = i & 255;
        float aq = 0.f, ak = 0.f, av = 0.f;
        for (int cp = 0; cp < C_; cp++) {
            float p = sP[(n << 8) + cp];
            aq += p * qw[(cp << 8) + c];
            ak += p * kw[(cp << 8) + c];
            av += p * vw[(cp << 8) + c];
        }
        sQ[i] = aq + qb_[c]; sK[i] = ak + kb_[c]; sV[i] = av + vb_[c];
    }
    __syncthreads();

    if (tid < N_ * N_) {
        int n = tid / N_, m = tid % N_;
        float s = 0.f;
        for (int c = 0; c < C_; c++) s += sQ[(n << 8) + c] * sK[(m << 8) + c];
        sAttn[tid] = s * SCALE_;
    }
    __syncthreads();
    if (tid < N_) {
        float mx = -1e30f;
        for (int m = 0; m < N_; m++) mx = fmaxf(mx, sAttn[tid * N_ + m]);
        float ssum = 0.f;
        for (int m = 0; m < N_; m++) { float e = expf(sAttn[tid * N_ + m] - mx); sAttn[tid * N_ + m] = e; ssum += e; }
        for (int m = 0; m < N_; m++) sAttn[tid * N_ + m] /= ssum;
    }
    __syncthreads();

    // P1 = LN(P + attn @ V)
    for (int n = 0; n < N_; n++) {
        float x = sP[(n << 8) + tid];
        for (int m = 0; m < N_; m++) x += sAttn[n * N_ + m] * sV[(m << 8) + tid];
        float y = block_ln_row(x, red, ls[tid], lb[tid], tid);
        sQ[(n << 8) + tid] = y;
        P1[(n << 8) + tid] = y;
    }
    __syncthreads();

    // Qc = P1 @ wq_w + wq_b
    for (int n = 0; n < N_; n++) {
        float a = 0.f;
        for (int cp = 0; cp < C_; cp++) a += sQ[(n << 8) + cp] * wq_w[(cp << 8) + tid];
        sQc[(n << 8) + tid] = a + wq_b[tid];
    }
    __syncthreads();

    // Qk[n][c] = sum_c' wk_w[c][c'] * Qc[n][c']  (padded to 16 rows with zeros)
    for (int n = 0; n < 16; n++) {
        float a = 0.f;
        if (n < N_) for (int cp = 0; cp < C_; cp++) a += wk_w[(tid << 8) + cp] * sQc[(n << 8) + cp];
        Qk[(n << 8) + tid] = a;
    }
    // qb[n] = scale * (Qc[n] . wk_b)
    for (int n = 0; n < N_; n++) {
        red[tid] = sQc[(n << 8) + tid] * wk_b[tid];
        __syncthreads();
        for (int off = 128; off > 0; off >>= 1) { if (tid < off) red[tid] += red[tid + off]; __syncthreads(); }
        if (tid == 0) qbo[n] = red[0] * SCALE_;
        __syncthreads();
    }
    if (tid >= N_ && tid < 16) qbo[tid] = 0.f;
}

// ---------- K1: logits[b,n,m] = scale*Fd[b,:,m].Qk[n] + qb[n]  (fp32 WMMA) ----------
// One wave per (b, 16-pixel tile). A = Qk (M=16 prompts x K=4), B = Fd slice (K=4 x N=16 pixels).
__global__ void k1_logits(const float* __restrict__ Fd, const float* __restrict__ Qk,
                          const float* __restrict__ qb, float* __restrict__ logits)
{
    int gt = blockIdx.x * blockDim.x + threadIdx.x;
    int wid = gt >> 5, lane = gt & 31;
    int b = wid >> 8, m0 = (wid & 255) << 4;
    int half = lane >> 4, l16 = lane & 15;

    const float* fdb = Fd + ((size_t)b * C_) * HW_ + m0 + l16;   // Fd[b][c][m0+l16]
    const float* qkp = Qk + (l16 << 8) + (half << 1);            // Qk[prompt=l16][c0 + 2*half + j]

    v8f acc = {0.f, 0.f, 0.f, 0.f, 0.f, 0.f, 0.f, 0.f};
    for (int c0 = 0; c0 < C_; c0 += 4) {
        v2f a;  a.x = qkp[c0];  a.y = qkp[c0 + 1];
        const float* f = fdb + (size_t)(c0 + (half << 1)) * HW_;
        v2f bb; bb.x = f[0];    bb.y = f[HW_];
        acc = __builtin_amdgcn_wmma_f32_16x16x4_f32(false, a, false, bb, (short)0, acc, false, false);
    }
    if (half == 0) {
#pragma unroll
        for (int g = 0; g < N_; ++g)   // D: vgpr g holds M=g (lanes 0-15), N=l16
            logits[((size_t)(b * N_ + g)) * HW_ + m0 + l16] = acc[g] * SCALE_ + qb[g];
    }
}

// ---------- K2: per-row (b,n) max and sum(exp) over HW ----------
__global__ void k2_rowstats(const float* __restrict__ logits,
                            float* __restrict__ mrow, float* __restrict__ srow)
{
    const int r = blockIdx.x;                 // r = b*N_ + n, 0..79
    const int tid = threadIdx.x;
    const float* row = logits + (size_t)r * HW_;
    __shared__ float red[256];
    float mx = -1e30f;
    for (int i = tid; i < HW_; i += 256) mx = fmaxf(mx, row[i]);
    red[tid] = mx; __syncthreads();
    for (int off = 128; off > 0; off >>= 1) { if (tid < off) red[tid] = fmaxf(red[tid], red[tid + off]); __syncthreads(); }
    mx = red[0]; __syncthreads();
    float s = 0.f;
    for (int i = tid; i < HW_; i += 256) s += expf(row[i] - mx);
    red[tid] = s; __syncthreads();
    for (int off = 128; off > 0; off >>= 1) { if (tid < off) red[tid] += red[tid + off]; __syncthreads(); }
    if (tid == 0) { mrow[r] = mx; srow[r] = red[0]; }
}

// ---------- K3: Sinkhorn (logK in registers) + BCE partial per batch ----------
__global__ void k3_sinkhorn(const float* __restrict__ logits, const float* __restrict__ mrow,
                            const float* __restrict__ srow, float* __restrict__ Lp)
{
    const int b = blockIdx.x, tid = threadIdx.x;
    const int wv = tid >> 5, ln = tid & 31;
    __shared__ float smax[8];
    __shared__ float ssum[8][N_];

    float lk[N_][16], lv[16], lu[N_], mK[N_];
    const float* base = logits + ((size_t)b * N_) * HW_ + tid * 16;
#pragma unroll
    for (int n = 0; n < N_; n++) {
#pragma unroll
        for (int j = 0; j < 16; j++) lk[n][j] = base[(size_t)n * HW_ + j] * (1.0f / EPS_);
        lu[n] = 0.f;
        mK[n] = mrow[b * N_ + n] * (1.0f / EPS_);
    }
#pragma unroll
    for (int j = 0; j < 16; j++) lv[j] = 0.f;

    const float log_mu = logf(1.0f / N_ + 1e-8f);
    const float log_nu = logf(1.0f / HW_ + 1e-8f);

    float diff = INFINITY;
    for (int iter = 0; iter < 100 && diff >= TOL_; ++iter) {
        // global max of lv (stability shift)
        float lm = lv[0];
#pragma unroll
        for (int j = 1; j < 16; j++) lm = fmaxf(lm, lv[j]);
        for (int o = 16; o; o >>= 1) lm = fmaxf(lm, __shfl_xor(lm, o, 32));
        if (ln == 0) smax[wv] = lm;
        __syncthreads();
        float lvmax = smax[0];
#pragma unroll
        for (int w = 1; w < 8; w++) lvmax = fmaxf(lvmax, smax[w]);

        // row LSE partials
        float ps[N_];
#pragma unroll
        for (int n = 0; n < N_; n++) {
            float sh = mK[n] + lvmax, s = 0.f;
#pragma unroll
            for (int j = 0; j < 16; j++) s += expf(lk[n][j] + lv[j] - sh);
            for (int o = 16; o; o >>= 1) s += __shfl_xor(s, o, 32);
            ps[n] = s;
        }
        if (ln == 0) {
#pragma unroll
            for (int n = 0; n < N_; n++) ssum[wv][n] = ps[n];
        }
        __syncthreads();

        // every thread computes identical new lu + diff (uniform control flow)
        float nd = 0.f;
#pragma unroll
        for (int n = 0; n < N_; n++) {
            float t = 0.f;
#pragma unroll
            for (int w = 0; w < 8; w++) t += ssum[w][n];
            float nlu = log_mu - (mK[n] + lvmax + logf(t));
            nd = fmaxf(nd, fabsf(nlu - lu[n]));
            lu[n] = nlu;
        }
        diff = nd;
        __syncthreads();

        // column update: LSE over only N_=5 values, fully local
#pragma unroll
        for (int j = 0; j < 16; j++) {
            float vs[N_]; float mx = lk[0][j] + lu[0]; vs[0] = mx;
#pragma unroll
            for (int n = 1; n < N_; n++) { vs[n] = lk[n][j] + lu[n]; mx = fmaxf(mx, vs[n]); }
            float s = 0.f;
#pragma unroll
            for (int n = 0; n < N_; n++) s += expf(vs[n] - mx);
            lv[j] = log_nu - (mx + logf(s));
        }
    }

    // BCE(T, M)
    float acc = 0.f;
#pragma unroll
    for (int n = 0; n < N_; n++) {
        float mr = mrow[b * N_ + n], isr = 1.0f / srow[b * N_ + n];
#pragma unroll
        for (int j = 0; j < 16; j++) {
            float lg = lk[n][j] * EPS_;
            float T  = fminf(fmaxf(expf(lu[n] + lk[n][j] + lv[j]), 0.f), 1.f);
            float M  = expf(lg - mr) * isr;
            float Mc = fminf(fmaxf(M, 1e-6f), 1.0f - 1e-6f);
            acc -= T * logf(Mc) + (1.0f - T) * logf(1.0f - Mc);
        }
    }
    for (int o = 16; o; o >>= 1) acc += __shfl_xor(acc, o, 32);
    if (ln == 0) smax[wv] = acc;
    __syncthreads();
    if (tid == 0) { float s = 0.f; for (int w = 0; w < 8; w++) s += smax[w]; Lp[b] = s; }
}

// ---------- K4: partial Aw over m-chunks (fp32 WMMA over K=m), 2048 waves ----------
// Awp[ch][b][n][c] = sum_{m in chunk ch} M[b,n,m] * Fd[b,c,m]
__global__ void k4_aw_partial(const float* __restrict__ Fd, const float* __restrict__ logits,
                              const float* __restrict__ mrow, const float* __restrict__ srow,
                              float* __restrict__ Awp)
{
    int gt = blockIdx.x * blockDim.x + threadIdx.x;
    int wid = gt >> 5, lane = gt & 31;
    int ch = wid & (MCH_ - 1);
    int ct = (wid >> 3) & 15;
    int b  = wid >> 7;
    int c0 = ct << 4;
    int m_lo = ch * (HW_ / MCH_), m_hi = m_lo + (HW_ / MCH_);
    int half = lane >> 4, l16 = lane & 15;

    int prompt = l16;
    bool pv = prompt < N_;
    int pr = pv ? prompt : 0;
    float mr  = mrow[b * N_ + pr];
    float isr = 1.0f / srow[b * N_ + pr];
    const float* lrow = logits + ((size_t)(b * N_ + pr)) * HW_ + (half << 1);       // A: M[prompt][m]
    const float* fdb  = Fd + ((size_t)(b * C_ + c0 + l16)) * HW_ + (half << 1);     // B: Fd^T[m][c]

    v8f acc = {0.f, 0.f, 0.f, 0.f, 0.f, 0.f, 0.f, 0.f};
    for (int m0 = m_lo; m0 < m_hi; m0 += 4) {
        float e0 = expf(lrow[m0]     - mr) * isr;
        float e1 = expf(lrow[m0 + 1] - mr) * isr;
        v2f a;  a.x = pv ? e0 : 0.f;  a.y = pv ? e1 : 0.f;
        v2f bb; bb.x = fdb[m0];       bb.y = fdb[m0 + 1];
        acc = __builtin_amdgcn_wmma_f32_16x16x4_f32(false, a, false, bb, (short)0, acc, false, false);
    }
    if (half == 0) {
#pragma unroll
        for (int g = 0; g < N_; ++g)
            Awp[(size_t)ch * (B_ * N_ * C_) + ((size_t)(b * N_ + g)) * C_ + c0 + l16] = acc[g];
    }
}

// ---------- K4b: reduce m-chunk partials ----------
__global__ void k4b_reduce(const float* __restrict__ Awp, float* __restrict__ Aw)
{
    int i = blockIdx.x * 256 + threadIdx.x;       // 0 .. B_*N_*C_-1
    if (i < B_ * N_ * C_) {
        float s = 0.f;
#pragma unroll
        for (int ch = 0; ch < MCH_; ch++) s += Awp[(size_t)ch * (B_ * N_ * C_) + i];
        Aw[i] = s;
    }
}

// ---------- K5: per-batch cross-attn output + LN + FFN + LN ----------
__global__ void k5_final(const float* __restrict__ P1, const float* __restrict__ Aw,
                         const float* __restrict__ wv_w, const float* __restrict__ wv_b,
                         const float* __restrict__ ca_s, const float* __restrict__ ca_b,
                         const float* __restrict__ w1,   const float* __restrict__ b1,
                         const float* __restrict__ w2,   const float* __restrict__ b2,
                         const float* __restrict__ fs,   const float* __restrict__ fb,
                         float* __restrict__ out)
{
    const int b = blockIdx.x, tid = threadIdx.x;
    __shared__ float sP2[N_ * C_], sh[N_ * E_], red[C_];

    for (int n = 0; n < N_; n++) {
        const float* aw = Aw + ((size_t)(b * N_ + n)) * C_;
        float s = 0.f;
        for (int cp = 0; cp < C_; cp++) s += aw[cp] * wv_w[(cp << 8) + tid];
        float x = P1[(n << 8) + tid] + s + wv_b[tid];
        sP2[(n << 8) + tid] = block_ln_row(x, red, ca_s[tid], ca_b[tid], tid);
    }
    __syncthreads();

    for (int n = 0; n < N_; n++) {
        for (int e = tid; e < E_; e += C_) {
            float s = 0.f;
            for (int c = 0; c < C_; c++) s += sP2[(n << 8) + c] * w1[c * E_ + e];
            s += b1[e];
            sh[n * E_ + e] = 0.5f * s * (1.0f + erff(s * 0.70710678118654752440f));  // exact GELU
        }
    }
    __syncthreads();

    for (int n = 0; n < N_; n++) {
        float s = 0.f;
        for (int e = 0; e < E_; e++) s += sh[n * E_ + e] * w2[(e << 8) + tid];
        float x = sP2[(n << 8) + tid] + s + b2[tid];
        out[((size_t)(b * N_ + n)) * C_ + tid] = block_ln_row(x, red, fs[tid], fb[tid], tid);
    }
}

// ---------- K6: final loss reduction ----------
__global__ void k6_loss(const float* __restrict__ Lp, float* __restrict__ out)
{
    if (threadIdx.x == 0 && blockIdx.x == 0) {
        float s = 0.f;
        for (int b = 0; b < B_; b++) s += Lp[b];
        out[B_ * N_ * C_] = s * (1.0f / ((float)B_ * N_ * HW_));
    }
}

extern "C" void kernel_launch(void* const* d_in, const int* in_sizes, int n_in,
                              void* d_out, int out_size, void* d_ws, size_t ws_size,
                              hipStream_t stream)
{
    const float* Fd      = (const float*)d_in[0];
    const float* prompts = (const float*)d_in[1];
    const float* sa_q_w  = (const float*)d_in[2];
    const float* sa_q_b  = (const float*)d_in[3];
    const float* sa_k_w  = (const float*)d_in[4];
    const float* sa_k_b  = (const float*)d_in[5];
    const float* sa_v_w  = (const float*)d_in[6];
    const float* sa_v_b  = (const float*)d_in[7];
    const float* sa_ln_s = (const float*)d_in[8];
    const float* sa_ln_b = (const float*)d_in[9];
    const float* wq_w    = (const float*)d_in[10];
    const float* wq_b    = (const float*)d_in[11];
    const float* wk_w    = (const float*)d_in[12];
    const float* wk_b    = (const float*)d_in[13];
    const float* wv_w    = (const float*)d_in[14];
    const float* wv_b    = (const float*)d_in[15];
    const float* ca_ln_s = (const float*)d_in[16];
    const float* ca_ln_b = (const float*)d_in[17];
    const float* ffn_w1  = (const float*)d_in[18];
    const float* ffn_b1  = (const float*)d_in[19];
    const float* ffn_w2  = (const float*)d_in[20];
    const float* ffn_b2  = (const float*)d_in[21];
    const float* ffn_ln_s= (const float*)d_in[22];
    const float* ffn_ln_b= (const float*)d_in[23];
    float* out = (float*)d_out;

    float* ws     = (float*)d_ws;
    float* P1     = ws;                // 1280
    float* Qk     = ws + 1280;         // 4096 (16 padded rows x 256)
    float* qb     = ws + 5376;         // 16
    float* mrow   = ws + 5392;         // 80
    float* srow   = ws + 5472;         // 80
    float* Lp     = ws + 5552;         // 16
    float* Aw     = ws + 5568;         // 20480
    float* logits = ws + 26048;        // 327680
    float* Awp    = ws + 353728;       // 163840  (total ~2.07 MB)

    k0_prompt_pipeline<<<1, 256, 0, stream>>>(prompts, sa_q_w, sa_q_b, sa_k_w, sa_k_b,
                                              sa_v_w, sa_v_b, sa_ln_s, sa_ln_b,
                                              wq_w, wq_b, wk_w, wk_b, P1, Qk, qb);
    k1_logits<<<1024, 128, 0, stream>>>(Fd, Qk, qb, logits);             // 4096 WMMA waves
    k2_rowstats<<<B_ * N_, 256, 0, stream>>>(logits, mrow, srow);
    k3_sinkhorn<<<B_, 256, 0, stream>>>(logits, mrow, srow, Lp);
    k4_aw_partial<<<512, 128, 0, stream>>>(Fd, logits, mrow, srow, Awp); // 2048 WMMA waves
    k4b_reduce<<<(B_ * N_ * C_ + 255) / 256, 256, 0, stream>>>(Awp, Aw);
    k5_final<<<B_, 256, 0, stream>>>(P1, Aw, wv_w, wv_b, ca_ln_s, ca_ln_b,
                                     ffn_w1, ffn_b1, ffn_w2, ffn_b2,
                                     ffn_ln_s, ffn_ln_b, out);
    k6_loss<<<1, 32, 0, stream>>>(Lp, out);
}